// GeoFNOT_1236950582159
// MI455X (gfx1250) — compile-verified
//
#include <hip/hip_runtime.h>
#include <math.h>

// ---------------------------------------------------------------------------
// GeoFNO forward for MI455X (gfx1250, wave32).
//  - f32 WMMA (v_wmma_f32_16x16x4_f32) for lift / pointwise-conv / proj1 GEMMs
//  - TDM (tensor_load_to_lds) DMAs the GEMM input tiles into LDS with
//    hardware padding (stride+1) for bank-conflict-free column reads,
//    synchronized with s_wait_tensorcnt.
//  - truncated DFT stages instead of full rfft2 (only 32x16 modes are used)
//  - spectral mix kernel is HBM-bound (268MB of weights, ~2 flop/byte):
//    coalesced streaming layout + global prefetch, batches fused so weights
//    are read exactly once.
// ---------------------------------------------------------------------------

typedef float v2f __attribute__((ext_vector_type(2)));
typedef float v8f __attribute__((ext_vector_type(8)));
typedef unsigned int u32x4 __attribute__((ext_vector_type(4)));
typedef int i32x8 __attribute__((ext_vector_type(8)));
typedef int i32x4 __attribute__((ext_vector_type(4)));

#define BB 2
#define NN 16384
#define WID 128
#define GG 4096
#define TWO_PI_OVER_64 0.09817477042468103f

__device__ __forceinline__ float gelu_f(float x) {
  float x3 = x * x * x;
  return 0.5f * x * (1.0f + tanhf(0.7978845608028654f * (x + 0.044715f * x3)));
}

// ---------------------------------------------------------------------------
// TDM: 2D tile DMA global->LDS with +1 DWORD padding after every row run.
//   tile_w elements per row (f32), tile_h rows, row stride stride_elems,
//   pad_code: 3 -> pad after 16 DWORDs, 6 -> pad after 128 DWORDs (1 DWORD).
// D# layout per CDNA5 ISA 8.3/8.4 (group0 128b, group1 256b; groups 2/3 NULL
// for 2D tiles). 6-arg builtin form (clang-23 / therock headers).
// ---------------------------------------------------------------------------
__device__ __forceinline__ void tdm_load_2d_padded(const float* gbase,
                                                   unsigned lds_byte,
                                                   unsigned tile_w, unsigned tile_h,
                                                   unsigned stride_elems,
                                                   unsigned tensor_h, int pad_code) {
  unsigned long long ga = (unsigned long long)(uintptr_t)gbase;
  u32x4 g0;
  g0[0] = 1u;                                   // count=1 valid descriptor
  g0[1] = lds_byte;                             // lds_addr (bytes)
  g0[2] = (unsigned)ga;                         // global_addr[31:0]
  g0[3] = (unsigned)((ga >> 32) & 0x01FFFFFFu)  // global_addr[56:32]
          | (2u << 30);                         // type = 2 ("image")
  i32x8 g1;
  g1[0] = (int)((2u << 16)                      // data_size = 4 bytes
                | (1u << 20)                    // pad_enable
                | ((unsigned)pad_code << 22));  // pad_interval; pad_amount=0 (1 DWORD)
  g1[1] = (int)((stride_elems & 0xFFFFu) << 16);          // tensor_dim0[15:0]
  g1[2] = (int)(((stride_elems >> 16) & 0xFFFFu)          // tensor_dim0[31:16]
                | ((tensor_h & 0xFFFFu) << 16));          // tensor_dim1[15:0]
  g1[3] = (int)(((tensor_h >> 16) & 0xFFFFu)              // tensor_dim1[31:16]
                | ((tile_w & 0xFFFFu) << 16));            // tile_dim0
  g1[4] = (int)(tile_h & 0xFFFFu);                        // tile_dim1 (tile_dim2=0)
  g1[5] = (int)stride_elems;                              // tensor_dim0_stride[31:0]
  g1[6] = 0;                                              // stride hi / dim1_stride
  g1[7] = 0;
  i32x4 gz4 = {0, 0, 0, 0};                               // groups 2/3: unused (2D)
  i32x8 gz8 = {0, 0, 0, 0, 0, 0, 0, 0};
  __builtin_amdgcn_tensor_load_to_lds(g0, g1, gz4, gz4, gz8, 0);
}

// ---------------------------------------------------------------------------
// 1) time embedding + concat -> xcat (B*N, 12)
// ---------------------------------------------------------------------------
__global__ void k_xcat(const float* __restrict__ nf, const float* __restrict__ tn,
                       float* __restrict__ xcat) {
  int t = blockIdx.x * blockDim.x + threadIdx.x;
  if (t >= BB * NN) return;
  int b = t >> 14;
  float tm = tn[b] * 1000.0f;
  float* row = xcat + (size_t)t * 12;
  const float* nr = nf + (size_t)t * 4;
#pragma unroll
  for (int c = 0; c < 4; ++c) row[c] = nr[c];
#pragma unroll
  for (int i = 0; i < 4; ++i) {
    float omega = expf(-(float)i * (6.907755278982137f / 4.0f)); // 1000^(-i/4)
    float ang = tm * omega;
    row[4 + i] = sinf(ang);
    row[8 + i] = cosf(ang);
  }
}

// ---------------------------------------------------------------------------
// 2) generic WMMA f32 GEMM:  C(M,128) = act(A(M,K) @ W(K,128) + bias)
//    block = 256 thr (8 waves); block owns 16 rows x all 128 cols.
//    K==128: A tile arrives via TDM DMA (padded stride 129); else manual.
// ---------------------------------------------------------------------------
template <int K, bool GELU>
__global__ void k_wmma_gemm(const float* __restrict__ A, const float* __restrict__ W,
                            const float* __restrict__ bias, float* __restrict__ C) {
  constexpr int KP = K + 1; // odd stride -> conflict-free column reads
  __shared__ float As[16 * KP];
  const int mbase = blockIdx.x * 16;
  const int tid = threadIdx.x;
  if constexpr (K == 128) {
    if (tid < 32) {
      tdm_load_2d_padded(A + (size_t)mbase * K, (unsigned)(uintptr_t)&As[0],
                         /*tile_w=*/128, /*tile_h=*/16, /*stride=*/128,
                         /*tensor_h=*/1u << 20, /*pad_code=*/6);
      __builtin_amdgcn_s_wait_tensorcnt((short)0);
    }
  } else {
    for (int idx = tid; idx < 16 * K; idx += 256) {
      int m = idx / K, k = idx - m * K;
      As[m * KP + k] = A[(size_t)(mbase + m) * K + k];
    }
  }
  __syncthreads();
  const int wave = tid >> 5, lane = tid & 31;
  const int nbase = wave * 16;
  const int nl = lane & 15;
  const int koff = (lane >= 16) ? 2 : 0;
  v8f acc = {};
#pragma unroll
  for (int kk = 0; kk < K; kk += 4) {
    v2f a, bb;
    a.x = As[nl * KP + kk + koff];
    a.y = As[nl * KP + kk + koff + 1];
    bb.x = W[(size_t)(kk + koff) * 128 + nbase + nl];
    bb.y = W[(size_t)(kk + koff + 1) * 128 + nbase + nl];
    acc = __builtin_amdgcn_wmma_f32_16x16x4_f32(false, a, false, bb, (short)0, acc,
                                                false, false);
  }
  const int mlo = (lane >= 16) ? 8 : 0;
#pragma unroll
  for (int r = 0; r < 8; ++r) {
    int m = mbase + r + mlo;
    int n = nbase + nl;
    float v = acc[r] + bias[n];
    if (GELU) v = gelu_f(v);
    C[(size_t)m * 128 + n] = v;
  }
}

// ---------------------------------------------------------------------------
// 3) deformation MLP -> latent (B*N, 2)
// ---------------------------------------------------------------------------
__global__ void k_defmlp(const float* __restrict__ coords,
                         const float* __restrict__ W0, const float* __restrict__ b0,
                         const float* __restrict__ W1, const float* __restrict__ b1,
                         const float* __restrict__ W2, const float* __restrict__ b2,
                         float* __restrict__ latent) {
  __shared__ float sW0[64], sb0[32], sW1[1024], sb1[32], sW2[64], sb2[2];
  int tid = threadIdx.x;
  for (int i = tid; i < 1024; i += 256) sW1[i] = W1[i];
  if (tid < 64) { sW0[tid] = W0[tid]; sW2[tid] = W2[tid]; }
  if (tid < 32) { sb0[tid] = b0[tid]; sb1[tid] = b1[tid]; }
  if (tid < 2) sb2[tid] = b2[tid];
  __syncthreads();
  int t = blockIdx.x * blockDim.x + tid;
  if (t >= BB * NN) return;
  float c0 = coords[(size_t)t * 2], c1 = coords[(size_t)t * 2 + 1];
  float h0[32], h1[32];
#pragma unroll
  for (int j = 0; j < 32; ++j)
    h0[j] = gelu_f(c0 * sW0[j] + c1 * sW0[32 + j] + sb0[j]);
#pragma unroll
  for (int j = 0; j < 32; ++j) {
    float a = sb1[j];
#pragma unroll
    for (int k = 0; k < 32; ++k) a += h0[k] * sW1[k * 32 + j];
    h1[j] = gelu_f(a);
  }
#pragma unroll
  for (int d = 0; d < 2; ++d) {
    float a = sb2[d];
#pragma unroll
    for (int k = 0; k < 32; ++k) a += h1[k] * sW2[k * 2 + d];
    latent[(size_t)t * 2 + d] = fminf(fmaxf(a, -1.0f), 1.0f);
  }
}

// ---------------------------------------------------------------------------
// 4) top-6 nearest nodes per grid center + IDW scatter -> G (B,128,4096)
//    one 256-thread block per (b,g)
// ---------------------------------------------------------------------------
__global__ void k_topk(const float* __restrict__ latent, const float* __restrict__ X,
                       float* __restrict__ Gout) {
  const float FINF = 3.0e38f;
  int b = blockIdx.x >> 12;
  int g = blockIdx.x & 4095;
  int gi = g >> 6, gj = g & 63;
  float cx = -1.0f + 2.0f * (float)gi / 63.0f;
  float cy = -1.0f + 2.0f * (float)gj / 63.0f;
  int tid = threadIdx.x;
  float d0 = FINF, d1 = FINF, d2 = FINF, d3 = FINF, d4 = FINF, d5 = FINF;
  int i0 = 0, i1 = 0, i2 = 0, i3 = 0, i4 = 0, i5 = 0;
  const float* latb = latent + (size_t)b * NN * 2;
  for (int n = tid; n < NN; n += 256) {
    float dx = cx - latb[n * 2];
    float dy = cy - latb[n * 2 + 1];
    float dist = sqrtf(fmaxf(dx * dx + dy * dy, 1e-12f));
    if (dist < d5) {
      if (dist < d4) { d5 = d4; i5 = i4;
        if (dist < d3) { d4 = d3; i4 = i3;
          if (dist < d2) { d3 = d2; i3 = i2;
            if (dist < d1) { d2 = d1; i2 = i1;
              if (dist < d0) { d1 = d0; i1 = i0; d0 = dist; i0 = n; }
              else           { d1 = dist; i1 = n; }
            } else { d2 = dist; i2 = n; }
          } else { d3 = dist; i3 = n; }
        } else { d4 = dist; i4 = n; }
      } else { d5 = dist; i5 = n; }
    }
  }
  __shared__ float sd[256 * 6];
  __shared__ int   si[256 * 6];
  __shared__ float rv[256];
  __shared__ int   rt[256];
  __shared__ float topd[6];
  __shared__ int   topi[6];
  sd[tid * 6 + 0] = d0; si[tid * 6 + 0] = i0;
  sd[tid * 6 + 1] = d1; si[tid * 6 + 1] = i1;
  sd[tid * 6 + 2] = d2; si[tid * 6 + 2] = i2;
  sd[tid * 6 + 3] = d3; si[tid * 6 + 3] = i3;
  sd[tid * 6 + 4] = d4; si[tid * 6 + 4] = i4;
  sd[tid * 6 + 5] = d5; si[tid * 6 + 5] = i5;
  int ptr = 0;
  for (int round = 0; round < 6; ++round) {
    rv[tid] = (ptr < 6) ? sd[tid * 6 + ptr] : FINF;
    rt[tid] = tid;
    __syncthreads();
    for (int s = 128; s > 0; s >>= 1) {
      if (tid < s && rv[tid + s] < rv[tid]) { rv[tid] = rv[tid + s]; rt[tid] = rt[tid + s]; }
      __syncthreads();
    }
    if (tid == rt[0]) {
      topd[round] = sd[tid * 6 + ptr];
      topi[round] = si[tid * 6 + ptr];
      ++ptr;
    }
    __syncthreads();
  }
  if (tid < 128) {
    float w[6], wsum = 0.0f;
#pragma unroll
    for (int k = 0; k < 6; ++k) { w[k] = 1.0f / (topd[k] + 1e-8f); wsum += w[k]; }
    float inv = 1.0f / wsum;
    float acc = 0.0f;
#pragma unroll
    for (int k = 0; k < 6; ++k)
      acc += (w[k] * inv) * X[((size_t)b * NN + topi[k]) * 128 + tid];
    Gout[((size_t)b * 128 + tid) * GG + g] = acc;
  }
}

// ---------------------------------------------------------------------------
// 5) truncated DFT stages (forward over y, forward over x) + inverse
// ---------------------------------------------------------------------------
__device__ __forceinline__ void tw_tables(float* tc, float* ts) {
  if (threadIdx.x < 64) {
    float ang = TWO_PI_OVER_64 * (float)threadIdx.x;
    tc[threadIdx.x] = cosf(ang);
    ts[threadIdx.x] = sinf(ang);
  }
  __syncthreads();
}

// G (B*128, 64, 64) -> FA (B*128, 64, 16) complex, e^{-i}
__global__ void k_dft_y(const float* __restrict__ G, float* __restrict__ FA) {
  __shared__ float tc[64], ts[64];
  tw_tables(tc, ts);
  int t = blockIdx.x * 256 + threadIdx.x; // 262144
  int ky = t & 15, x = (t >> 4) & 63, bi = t >> 10;
  const float* row = G + ((size_t)bi * 64 + x) * 64;
  float re = 0.0f, im = 0.0f;
#pragma unroll 8
  for (int y = 0; y < 64; ++y) {
    float v = row[y];
    int k = (ky * y) & 63;
    re += v * tc[k];
    im -= v * ts[k];
  }
  FA[(size_t)t * 2] = re;
  FA[(size_t)t * 2 + 1] = im;
}

// FA -> XF (B*128, 32, 16) complex; kx = {0..15, 48..63}
__global__ void k_dft_x(const float* __restrict__ FA, float* __restrict__ XF) {
  __shared__ float tc[64], ts[64];
  tw_tables(tc, ts);
  int t = blockIdx.x * 256 + threadIdx.x; // 131072
  int ky = t & 15, kxi = (t >> 4) & 31, bi = t >> 9;
  int kx = (kxi < 16) ? kxi : kxi + 32;
  float re = 0.0f, im = 0.0f;
#pragma unroll 8
  for (int x = 0; x < 64; ++x) {
    size_t ai = (((size_t)bi * 64 + x) * 16 + ky) * 2;
    float ar = FA[ai], aim = FA[ai + 1];
    int k = (kx * x) & 63;
    float c = tc[k], s = ts[k];
    re += ar * c + aim * s;
    im += aim * c - ar * s;
  }
  XF[(size_t)t * 2] = re;
  XF[(size_t)t * 2 + 1] = im;
}

// spectral mix: YF[b,o,m] = sum_i XF[b,i,m] * (w_re + i w_im)[i,o,m]
// HBM-bound: weights streamed once, both batches fused.
__global__ void k_specmul(const float* __restrict__ XF,
                          const float* __restrict__ w1re, const float* __restrict__ w1im,
                          const float* __restrict__ w2re, const float* __restrict__ w2im,
                          float* __restrict__ YF) {
  int t = blockIdx.x * 256 + threadIdx.x; // 65536
  int ky = t & 15, kxi = (t >> 4) & 31, o = t >> 9;
  int xw = (kxi < 16) ? kxi : (kxi - 16);
  const float* wre = (kxi < 16) ? w1re : w2re;
  const float* wim = (kxi < 16) ? w1im : w2im;
  int mode = xw * 16 + ky;
  float a0r = 0.f, a0i = 0.f, a1r = 0.f, a1i = 0.f;
  for (int i = 0; i < 128; ++i) {
    size_t widx = (size_t)(i * 128 + o) * 256 + mode;
    float wr = wre[widx], wi = wim[widx];
    __builtin_prefetch(&wre[widx + 32768]);
    __builtin_prefetch(&wim[widx + 32768]);
    size_t x0 = ((size_t)i * 32 + kxi) * 32 + ky * 2;
    size_t x1 = ((size_t)(128 + i) * 32 + kxi) * 32 + ky * 2;
    float xr0 = XF[x0], xi0 = XF[x0 + 1];
    float xr1 = XF[x1], xi1 = XF[x1 + 1];
    a0r += xr0 * wr - xi0 * wi; a0i += xr0 * wi + xi0 * wr;
    a1r += xr1 * wr - xi1 * wi; a1i += xr1 * wi + xi1 * wr;
  }
  size_t y0 = ((size_t)o * 32 + kxi) * 32 + ky * 2;
  size_t y1 = ((size_t)(128 + o) * 32 + kxi) * 32 + ky * 2;
  YF[y0] = a0r; YF[y0 + 1] = a0i;
  YF[y1] = a1r; YF[y1 + 1] = a1i;
}

// inverse over x (e^{+i}): YF -> TT (B*128, 64, 16) complex
__global__ void k_idft_x(const float* __restrict__ YF, float* __restrict__ TT) {
  __shared__ float tc[64], ts[64];
  tw_tables(tc, ts);
  int t = blockIdx.x * 256 + threadIdx.x; // 262144
  int ky = t & 15, x = (t >> 4) & 63, bo = t >> 10;
  float re = 0.0f, im = 0.0f;
#pragma unroll 8
  for (int kxi = 0; kxi < 32; ++kxi) {
    int kx = (kxi < 16) ? kxi : kxi + 32;
    size_t yi = (((size_t)bo * 32 + kxi) * 16 + ky) * 2;
    float yr = YF[yi], yim = YF[yi + 1];
    int k = (kx * x) & 63;
    float c = tc[k], s = ts[k];
    re += yr * c - yim * s;
    im += yr * s + yim * c;
  }
  TT[(size_t)t * 2] = re;
  TT[(size_t)t * 2 + 1] = im;
}

// inverse rfft over y: TT -> spec (B*128, 64, 64), norm 1/4096
__global__ void k_idft_y(const float* __restrict__ TT, float* __restrict__ spec) {
  __shared__ float tc[64], ts[64];
  tw_tables(tc, ts);
  int t = blockIdx.x * 256 + threadIdx.x; // 1048576
  int y = t & 63, x = (t >> 6) & 63, bo = t >> 12;
  size_t base = ((size_t)bo * 64 + x) * 16;
  float acc = TT[base * 2]; // ky = 0 (real part)
#pragma unroll
  for (int ky = 1; ky < 16; ++ky) {
    float tr = TT[(base + ky) * 2];
    float ti = TT[(base + ky) * 2 + 1];
    int k = (ky * y) & 63;
    acc += 2.0f * (tr * tc[k] - ti * ts[k]);
  }
  spec[t] = acc * (1.0f / 4096.0f);
}

// ---------------------------------------------------------------------------
// 6) pointwise conv (WMMA): Gn(b,o,s) = gelu(spec + bias[o] + sum_i W[o,i]G(b,i,s))
//    block: 256 thr, owns 16 spatial cols x all 128 out-channels.
//    G tile (128 rows x 16 f32, row stride 4096) DMA'd into LDS by the TDM
//    with 1-DWORD padding every 16 DWORDs -> LDS stride 17, conflict-free.
// ---------------------------------------------------------------------------
__global__ void k_pw(const float* __restrict__ G, const float* __restrict__ W,
                     const float* __restrict__ bias, const float* __restrict__ spec,
                     float* __restrict__ Gn) {
  int b = blockIdx.x >> 8;
  int sbase = (blockIdx.x & 255) * 16;
  __shared__ float Gs[128 * 17];
  int tid = threadIdx.x;
  if (tid < 32) {
    tdm_load_2d_padded(G + (size_t)b * 128 * GG + sbase, (unsigned)(uintptr_t)&Gs[0],
                       /*tile_w=*/16, /*tile_h=*/128, /*stride=*/GG,
                       /*tensor_h=*/128, /*pad_code=*/3);
    __builtin_amdgcn_s_wait_tensorcnt((short)0);
  }
  __syncthreads();
  const int wave = tid >> 5, lane = tid & 31;
  const int obase = wave * 16;
  const int nl = lane & 15;
  const int koff = (lane >= 16) ? 2 : 0;
  v8f acc = {};
#pragma unroll 8
  for (int kk = 0; kk < 128; kk += 4) {
    v2f a, bb;
    a.x = W[(size_t)(obase + nl) * 128 + kk + koff];
    a.y = W[(size_t)(obase + nl) * 128 + kk + koff + 1];
    bb.x = Gs[(kk + koff) * 17 + nl];
    bb.y = Gs[(kk + koff + 1) * 17 + nl];
    acc = __builtin_amdgcn_wmma_f32_16x16x4_f32(false, a, false, bb, (short)0, acc,
                                                false, false);
  }
  const int ohi = (lane >= 16) ? 8 : 0;
#pragma unroll
  for (int r = 0; r < 8; ++r) {
    int o = obase + r + ohi;
    size_t off = ((size_t)b * 128 + o) * GG + sbase + nl;
    Gn[off] = gelu_f(acc[r] + bias[o] + spec[off]);
  }
}

// ---------------------------------------------------------------------------
// 7) grid transpose (b,w,s) -> (b,s,w), then bilinear sample, then proj2
// ---------------------------------------------------------------------------
__global__ void k_transpose(const float* __restrict__ Gin, float* __restrict__ GT) {
  int t = blockIdx.x * 256 + threadIdx.x; // 1048576
  int w = t & 127, s = (t >> 7) & 4095, b = t >> 19;
  GT[((size_t)b * GG + s) * 128 + w] = Gin[((size_t)b * 128 + w) * GG + s];
}

__global__ void k_sample(const float* __restrict__ GT, const float* __restrict__ latent,
                         float* __restrict__ nodes) {
  int bn = blockIdx.x; // B*N
  int b = bn >> 14;
  int w = threadIdx.x; // 128
  float lx = latent[(size_t)bn * 2], ly = latent[(size_t)bn * 2 + 1];
  float fi = fminf(fmaxf((lx + 1.0f) * 0.5f * 63.0f, 0.0f), 63.0f);
  float fj = fminf(fmaxf((ly + 1.0f) * 0.5f * 63.0f, 0.0f), 63.0f);
  int i0 = (int)floorf(fi); i0 = i0 < 0 ? 0 : (i0 > 62 ? 62 : i0);
  int j0 = (int)floorf(fj); j0 = j0 < 0 ? 0 : (j0 > 62 ? 62 : j0);
  float ti = fi - (float)i0, tj = fj - (float)j0;
  const float* gb = GT + (size_t)b * GG * 128;
  const float* p00 = gb + (size_t)(i0 * 64 + j0) * 128;
  const float* p01 = p00 + 128;        // j0+1
  const float* p10 = p00 + 64 * 128;   // i0+1
  const float* p11 = p10 + 128;
  float v = p00[w] * (1.0f - ti) * (1.0f - tj) + p01[w] * (1.0f - ti) * tj +
            p10[w] * ti * (1.0f - tj) + p11[w] * ti * tj;
  nodes[(size_t)bn * 128 + w] = v;
}

__global__ void k_proj2(const float* __restrict__ H, const float* __restrict__ W,
                        const float* __restrict__ bias, float* __restrict__ out) {
  int t = blockIdx.x * 256 + threadIdx.x; // 131072
  int c = t & 3, row = t >> 2;
  float acc = bias[c];
  const float* hr = H + (size_t)row * 128;
#pragma unroll 8
  for (int k = 0; k < 128; ++k) acc += hr[k] * W[k * 4 + c];
  out[(size_t)row * 4 + c] = acc;
}

// ---------------------------------------------------------------------------
// host launcher
// ---------------------------------------------------------------------------
extern "C" void kernel_launch(void* const* d_in, const int* in_sizes, int n_in,
                              void* d_out, int out_size, void* d_ws, size_t ws_size,
                              hipStream_t stream) {
  (void)in_sizes; (void)n_in; (void)out_size; (void)ws_size;
  const float* nf     = (const float*)d_in[0];
  const float* coords = (const float*)d_in[1];
  const float* tn     = (const float*)d_in[2];
  const float* defW0  = (const float*)d_in[3];
  const float* defb0  = (const float*)d_in[4];
  const float* defW1  = (const float*)d_in[5];
  const float* defb1  = (const float*)d_in[6];
  const float* defW2  = (const float*)d_in[7];
  const float* defb2  = (const float*)d_in[8];
  const float* liftW  = (const float*)d_in[9];
  const float* liftb  = (const float*)d_in[10];
  const float* w1re   = (const float*)d_in[11];
  const float* w1im   = (const float*)d_in[12];
  const float* w2re   = (const float*)d_in[13];
  const float* w2im   = (const float*)d_in[14];
  const float* pwW    = (const float*)d_in[15];
  const float* pwb    = (const float*)d_in[16];
  const float* p1W    = (const float*)d_in[17];
  const float* p1b    = (const float*)d_in[18];
  const float* p2W    = (const float*)d_in[19];
  const float* p2b    = (const float*)d_in[20];
  float* out = (float*)d_out;

  float* ws = (float*)d_ws;
  // float offsets (total ~54.3 MB)
  float* X    = ws;                   // 4,194,304  (x; later reused as nodes)
  float* H    = ws + 4194304;         // 4,194,304  (proj1 out)
  float* Gcur = ws + 8388608;         // 1,048,576
  float* Gsp  = ws + 9437184;         // 1,048,576  (spec; later reused as GT)
  float* Gnx  = ws + 10485760;        // 1,048,576
  float* FA   = ws + 11534336;        //   524,288
  float* XF   = ws + 12058624;        //   262,144
  float* YF   = ws + 12320768;        //   262,144
  float* TT   = ws + 12582912;        //   524,288
  float* XC   = ws + 13107200;        //   393,216
  float* LAT  = ws + 13500416;        //    65,536

  // lift path
  k_xcat<<<128, 256, 0, stream>>>(nf, tn, XC);
  k_wmma_gemm<12, false><<<2048, 256, 0, stream>>>(XC, liftW, liftb, X);
  // deformation MLP
  k_defmlp<<<128, 256, 0, stream>>>(coords, defW0, defb0, defW1, defb1, defW2, defb2, LAT);
  // geometry -> grid
  k_topk<<<8192, 256, 0, stream>>>(LAT, X, Gcur);

  // FNO layers
  for (int d = 0; d < 4; ++d) {
    const size_t wlay = (size_t)d * 128 * 128 * 256;
    k_dft_y<<<1024, 256, 0, stream>>>(Gcur, FA);
    k_dft_x<<<512, 256, 0, stream>>>(FA, XF);
    k_specmul<<<256, 256, 0, stream>>>(XF, w1re + wlay, w1im + wlay,
                                       w2re + wlay, w2im + wlay, YF);
    k_idft_x<<<1024, 256, 0, stream>>>(YF, TT);
    k_idft_y<<<4096, 256, 0, stream>>>(TT, Gsp);
    k_pw<<<512, 256, 0, stream>>>(Gcur, pwW + (size_t)d * 16384, pwb + (size_t)d * 128,
                                  Gsp, Gnx);
    float* tmp = Gcur; Gcur = Gnx; Gnx = tmp;
  }

  // grid -> points -> projection
  k_transpose<<<4096, 256, 0, stream>>>(Gcur, Gsp);      // Gsp now holds GT
  k_sample<<<32768, 128, 0, stream>>>(Gsp, LAT, X);      // X now holds nodes
  k_wmma_gemm<128, true><<<2048, 256, 0, stream>>>(X, p1W, p1b, H);
  k_proj2<<<512, 256, 0, stream>>>(H, p2W, p2b, out);
}